// CostVolume_47433618817551
// MI455X (gfx1250) — compile-verified
//
#include <hip/hip_runtime.h>

typedef __attribute__((ext_vector_type(2))) float v2f;
typedef __attribute__((ext_vector_type(8))) float v8f;

#define Bsz   4
#define Hsz   256
#define Wsz   512
#define Csz   64
#define Dmax  64
#define CHUNK 64       // w-values per workgroup
#define LSTRIDE 68     // padded LDS row stride (floats): conflict-free ds_load_b64

__global__ __launch_bounds__(128) void cost_volume_wmma(
    const float* __restrict__ left, const float* __restrict__ right,
    float* __restrict__ out)
{
    __shared__ float Ls[CHUNK * LSTRIDE];          // 64 rows  x 64ch (padded)
    __shared__ float Rs[(CHUNK + 64) * LSTRIDE];   // 128 rows x 64ch (padded)

    const int tid   = threadIdx.x;
    const int chunk = blockIdx.x & 7;              // Wsz/CHUNK = 8 chunks
    const int bh    = blockIdx.x >> 3;             // b*Hsz + h
    const int wc0   = chunk * CHUNK;
    const size_t rowbase = (size_t)bh * Wsz;       // (b,h) plane offset in w-units

    // ---- stage L chunk: 64 rows x 16 float4 = 1024 vec4 loads ----
    const float4* Lg = (const float4*)(left + (rowbase + wc0) * Csz);
    #pragma unroll
    for (int i = 0; i < 8; ++i) {
        int idx = tid + i * 128;
        int row = idx >> 4, c4 = idx & 15;
        float4 v = Lg[row * 16 + c4];
        float* dst = &Ls[row * LSTRIDE + c4 * 4];
        dst[0] = v.x; dst[1] = v.y; dst[2] = v.z; dst[3] = v.w;
    }
    // ---- stage R chunk: rows j = wc0-64 .. wc0+63 (clamped to >=0) ----
    #pragma unroll
    for (int i = 0; i < 16; ++i) {
        int idx = tid + i * 128;
        int row = idx >> 4, c4 = idx & 15;
        int j = wc0 - 64 + row; if (j < 0) j = 0;  // garbage cols discarded at store
        const float4* Rg = (const float4*)(right + (rowbase + j) * Csz);
        float4 v = Rg[c4];
        float* dst = &Rs[row * LSTRIDE + c4 * 4];
        dst[0] = v.x; dst[1] = v.y; dst[2] = v.z; dst[3] = v.w;
    }
    __syncthreads();

    const int wave = tid >> 5;       // 4 waves, one 16-row w-tile each
    const int lane = tid & 31;
    const int n16  = lane & 15;      // N (B/C/D) or M (A)
    const int khi  = lane >> 4;      // A/B: K pair {0,1} vs {2,3}; C/D: M offset 8

    v8f acc[5] = {};                 // 5 band tiles: j0 = w0-64+16t

    const int aRow = wave * 16 + n16;
    #pragma unroll
    for (int kk = 0; kk < 16; ++kk) {            // K = 64 in steps of 4
        const int k = kk * 4 + khi * 2;
        v2f a = *(const v2f*)&Ls[aRow * LSTRIDE + k];
        #pragma unroll
        for (int t = 0; t < 5; ++t) {
            const int bRow = wave * 16 + t * 16 + n16;   // (j - (wc0-64))
            v2f b = *(const v2f*)&Rs[bRow * LSTRIDE + k];
            acc[t] = __builtin_amdgcn_wmma_f32_16x16x4_f32(
                         /*neg_a=*/false, a, /*neg_b=*/false, b,
                         /*c_mod=*/(short)0, acc[t],
                         /*reuse_a=*/false, /*reuse_b=*/false);
        }
    }

    // ---- scatter band back to (w, d): d = 63 - 16t + M - N ----
    const int w0 = wc0 + wave * 16;
    float* outp = out + (size_t)bh * Wsz * Dmax;
    #pragma unroll
    for (int t = 0; t < 5; ++t) {
        #pragma unroll
        for (int r = 0; r < 8; ++r) {
            const int M = r + khi * 8;
            const int w = w0 + M;
            const int d = (63 - 16 * t) + M - n16;
            if (d >= 0 && d < Dmax) {
                const int j = w - d - 1;
                const float val = (j >= 0) ? acc[t][r] * 0.015625f : 0.0f;
                outp[(size_t)w * Dmax + d] = val;
            }
        }
    }
}

extern "C" void kernel_launch(void* const* d_in, const int* in_sizes, int n_in,
                              void* d_out, int out_size, void* d_ws, size_t ws_size,
                              hipStream_t stream) {
    (void)in_sizes; (void)n_in; (void)out_size; (void)d_ws; (void)ws_size;
    const float* left  = (const float*)d_in[0];
    const float* right = (const float*)d_in[1];
    float* out = (float*)d_out;
    dim3 grid(Bsz * Hsz * (Wsz / CHUNK));   // 8192 workgroups
    cost_volume_wmma<<<grid, 128, 0, stream>>>(left, right, out);
}